// DetectionLoss_52673478918915
// MI455X (gfx1250) — compile-verified
//
#include <hip/hip_runtime.h>
#include <math.h>

// ---------------- problem constants (match reference) ----------------
#define BQ     32            // batch
#define GQ     64            // GTs per image
#define NCQ    6             // classes
#define RB     17            // REG_MAX+1 bins
#define NL0    6400          // 80x80 (stride 8)
#define NL1    1600          // 40x40 (stride 16)
#define NL2    400           // 20x20 (stride 32)
#define NANCH  8400
#define CHN    74
#define TOPKQ  13
#define CHUNKS 33            // ceil(8400/256)

typedef float v2f __attribute__((ext_vector_type(2)));
typedef float v8f __attribute__((ext_vector_type(8)));

// ---------------- exact 32-lane fp32 sum via V_WMMA_F32_16X16X4_F32 ----------------
// A holds one lane value (second A reg = 0), B = all ones. D[m,n] = sum_k A[m,k].
// Lane l's 8 D VGPRs cover 8 distinct rows; lanes 0-15 cover rows 0-7, lanes 16-31
// rows 8-15, so lane-sum + xor-16 add = total of all 32 lane values regardless of
// the exact (m,k) striping. Requires EXEC all-ones (all callers keep full blocks).
__device__ __forceinline__ float wave_sum32(float x) {
  v2f a; a[0] = x;   a[1] = 0.0f;
  v2f b; b[0] = 1.f; b[1] = 1.0f;
  v8f c = {};
  v8f d = __builtin_amdgcn_wmma_f32_16x16x4_f32(false, a, false, b, (short)0, c, false, false);
  float s = d[0] + d[1] + d[2] + d[3] + d[4] + d[5] + d[6] + d[7];
  s += __shfl_xor(s, 16, 32);
  return s;
}

__device__ __forceinline__ float block_sum256(float v, float* sm8) {
  float w = wave_sum32(v);
  int lane = threadIdx.x & 31, wid = threadIdx.x >> 5;
  __syncthreads();                 // protect previous contents of sm8
  if (lane == 0) sm8[wid] = w;
  __syncthreads();
  float t = 0.f;
  if (threadIdx.x == 0) {
#pragma unroll
    for (int i = 0; i < 8; ++i) t += sm8[i];
  }
  return t;                        // valid on thread 0
}

__device__ __forceinline__ void anchor_of(int n, int& level, int& pos, int& S,
                                          float& ax, float& ay, float& st) {
  if (n < NL0)            { level = 0; pos = n;           S = NL0; st = 8.f;
                            ax = (float)(pos % 80) + 0.5f; ay = (float)(pos / 80) + 0.5f; }
  else if (n < NL0 + NL1) { level = 1; pos = n - NL0;     S = NL1; st = 16.f;
                            ax = (float)(pos % 40) + 0.5f; ay = (float)(pos / 40) + 0.5f; }
  else                    { level = 2; pos = n - NL0-NL1; S = NL2; st = 32.f;
                            ax = (float)(pos % 20) + 0.5f; ay = (float)(pos / 20) + 0.5f; }
}

__device__ __forceinline__ float iou_cxcywh(float px, float py, float pw, float ph,
                                            float gx, float gy, float gw, float gh) {
  float p_x1 = px - pw*0.5f, p_y1 = py - ph*0.5f, p_x2 = px + pw*0.5f, p_y2 = py + ph*0.5f;
  float g_x1 = gx - gw*0.5f, g_y1 = gy - gh*0.5f, g_x2 = gx + gw*0.5f, g_y2 = gy + gh*0.5f;
  float iw = fmaxf(fminf(p_x2, g_x2) - fmaxf(p_x1, g_x1), 0.f);
  float ih = fmaxf(fminf(p_y2, g_y2) - fmaxf(p_y1, g_y1), 0.f);
  float inter = iw * ih;
  float uni = pw*ph + gw*gh - inter + 1e-7f;
  return inter / uni;
}

// ---------------- kernel 1: decode DFL expectation + stash cls logits ----------------
__global__ void k_predict(const float* __restrict__ f0, const float* __restrict__ f1,
                          const float* __restrict__ f2,
                          float* __restrict__ pb, float* __restrict__ pcls) {
  int idx = blockIdx.x * blockDim.x + threadIdx.x;
  if (idx >= BQ * NANCH) return;
  int b = idx / NANCH, n = idx % NANCH;
  int level, pos, S; float ax, ay, st;
  anchor_of(n, level, pos, S, ax, ay, st);
  const float* f = (level == 0) ? f0 : (level == 1) ? f1 : f2;
  const float* base = f + (size_t)b * CHN * S + pos;
  float ltrb[4];
#pragma unroll
  for (int sd = 0; sd < 4; ++sd) {
    float v[RB]; float m = -1e30f;
#pragma unroll
    for (int k = 0; k < RB; ++k) { v[k] = base[(size_t)(sd*RB + k) * S]; m = fmaxf(m, v[k]); }
    float se = 0.f, ex = 0.f;
#pragma unroll
    for (int k = 0; k < RB; ++k) { float e = __expf(v[k] - m); se += e; ex += e * (float)k; }
    ltrb[sd] = ex / se;
  }
  float x1 = ax - ltrb[0], y1 = ay - ltrb[1], x2 = ax + ltrb[2], y2 = ay + ltrb[3];
  pb[(size_t)idx*4 + 0] = (x1 + x2) * 0.5f * st;
  pb[(size_t)idx*4 + 1] = (y1 + y2) * 0.5f * st;
  pb[(size_t)idx*4 + 2] = (x2 - x1) * st;
  pb[(size_t)idx*4 + 3] = (y2 - y1) * st;
#pragma unroll
  for (int cc = 0; cc < NCQ; ++cc) pcls[(size_t)idx*NCQ + cc] = base[(size_t)(68 + cc) * S];
}

// ---------------- kernel 2: per-(b,gt) 13th-largest alignment threshold ----------------
__global__ void k_topk(const float* __restrict__ tgt, const float* __restrict__ pb,
                       const float* __restrict__ pcls, float* __restrict__ kth) {
  int bm = blockIdx.x;            // b*GQ + m
  int b  = bm / GQ;
  const float* tr = tgt + (size_t)bm * 6;
  float gcls = tr[1];
  float gx = tr[2]*640.f, gy = tr[3]*640.f, gw = tr[4]*640.f, gh = tr[5]*640.f;
  int ci = (int)gcls; if (ci > NCQ-1) ci = NCQ-1; if (ci < 0) ci = 0;
  float g_x1 = gx - gw*0.5f, g_y1 = gy - gh*0.5f, g_x2 = gx + gw*0.5f, g_y2 = gy + gh*0.5f;

  float t13[TOPKQ];
#pragma unroll
  for (int i = 0; i < TOPKQ; ++i) t13[i] = -1.f;

  for (int n = threadIdx.x; n < NANCH; n += blockDim.x) {
    int level, pos, S; float ax, ay, st;
    anchor_of(n, level, pos, S, ax, ay, st);
    float apx = ax * st, apy = ay * st;
    bool ing = (apx > g_x1) && (apx < g_x2) && (apy > g_y1) && (apy < g_y2);
    float a = 0.f;
    if (ing) {
      size_t pidx = (size_t)b * NANCH + n;
      float px = pb[pidx*4], py = pb[pidx*4+1], pw = pb[pidx*4+2], ph = pb[pidx*4+3];
      float iou = fmaxf(iou_cxcywh(px, py, pw, ph, gx, gy, gw, gh), 1e-9f);
      float sc  = fmaxf(1.f / (1.f + __expf(-pcls[pidx*NCQ + ci])), 1e-9f);
      float i2 = iou * iou;
      a = sqrtf(sc) * i2 * i2 * i2;       // cls^0.5 * iou^6, masked by in_gts
    }
    if (a > t13[TOPKQ-1]) {
      int p = TOPKQ-1;
      while (p > 0 && t13[p-1] < a) { t13[p] = t13[p-1]; --p; }
      t13[p] = a;
    }
  }
  __shared__ float sm[128 * TOPKQ];
#pragma unroll
  for (int i = 0; i < TOPKQ; ++i) sm[threadIdx.x * TOPKQ + i] = t13[i];
  __syncthreads();
  if (threadIdx.x == 0) {
    float g13[TOPKQ];
#pragma unroll
    for (int i = 0; i < TOPKQ; ++i) g13[i] = -1.f;
    for (int j = 0; j < 128 * TOPKQ; ++j) {
      float a = sm[j];
      if (a > g13[TOPKQ-1]) {
        int p = TOPKQ-1;
        while (p > 0 && g13[p-1] < a) { g13[p] = g13[p-1]; --p; }
        g13[p] = a;
      }
    }
    kth[bm] = g13[TOPKQ-1];       // 13th largest align for this (b, gt)
  }
}

// ---------------- kernel 3: per-anchor assignment ----------------
__global__ void k_assign(const float* __restrict__ tgt, const float* __restrict__ pb,
                         const float* __restrict__ pcls, const float* __restrict__ kth,
                         float* __restrict__ tbb, float* __restrict__ apa,
                         int* __restrict__ tlab, int* __restrict__ fgm,
                         unsigned* __restrict__ mxbits, float* __restrict__ acc) {
  int b = blockIdx.x / CHUNKS;
  int n = (blockIdx.x % CHUNKS) * 256 + threadIdx.x;
  bool valid = n < NANCH;

  __shared__ float sgt[GQ * 5];   // cls, cx, cy, w, h (px)
  __shared__ float skth[GQ];
  __shared__ float sm8[8];
  if (threadIdx.x < GQ) {
    const float* tr = tgt + (size_t)(b * GQ + threadIdx.x) * 6;
    sgt[threadIdx.x*5 + 0] = tr[1];
    sgt[threadIdx.x*5 + 1] = tr[2] * 640.f;
    sgt[threadIdx.x*5 + 2] = tr[3] * 640.f;
    sgt[threadIdx.x*5 + 3] = tr[4] * 640.f;
    sgt[threadIdx.x*5 + 4] = tr[5] * 640.f;
    skth[threadIdx.x] = kth[b * GQ + threadIdx.x];
  }
  __syncthreads();

  size_t pidx = (size_t)b * NANCH + (valid ? n : 0);
  int level, pos, S; float ax, ay, st;
  anchor_of(valid ? n : 0, level, pos, S, ax, ay, st);
  float apx = ax * st, apy = ay * st;
  float px = pb[pidx*4], py = pb[pidx*4+1], pw = pb[pidx*4+2], ph = pb[pidx*4+3];
  float sc[NCQ];
#pragma unroll
  for (int cc = 0; cc < NCQ; ++cc) sc[cc] = 1.f / (1.f + __expf(-pcls[pidx*NCQ + cc]));

  bool fg = false; int best = 0; float bestv = -1.f, balign = 0.f;
  for (int m = 0; m < GQ; ++m) {
    float gcls = sgt[m*5], gx = sgt[m*5+1], gy = sgt[m*5+2], gw = sgt[m*5+3], gh = sgt[m*5+4];
    float g_x1 = gx - gw*0.5f, g_y1 = gy - gh*0.5f, g_x2 = gx + gw*0.5f, g_y2 = gy + gh*0.5f;
    bool ing = (apx > g_x1) && (apx < g_x2) && (apy > g_y1) && (apy < g_y2);
    float iou = iou_cxcywh(px, py, pw, ph, gx, gy, gw, gh);
    float a = 0.f;
    if (ing) {
      int ciq = (int)gcls; if (ciq > NCQ-1) ciq = NCQ-1; if (ciq < 0) ciq = 0;
      float s = fmaxf(sc[ciq], 1e-9f);
      float i = fmaxf(iou, 1e-9f);
      float i2 = i * i;
      a = sqrtf(s) * i2 * i2 * i2;
    }
    bool mpos = (a > 0.f) && (a >= skth[m]);        // is_topk & in_gts
    if (mpos) fg = true;
    float v = mpos ? iou : 0.f;
    if (v > bestv) { bestv = v; best = m; balign = a; }   // first-max tie-break
  }
  if (!valid) fg = false;
  if (valid) {
    fgm[pidx]    = fg ? 1 : 0;
    tlab[pidx]   = (int)sgt[best*5];
    tbb[pidx*4]  = sgt[best*5+1]; tbb[pidx*4+1] = sgt[best*5+2];
    tbb[pidx*4+2]= sgt[best*5+3]; tbb[pidx*4+3] = sgt[best*5+4];
    apa[pidx]    = balign;
  }
  // num_pos via WMMA-backed block sum
  float cnt = block_sum256((valid && fg) ? 1.f : 0.f, sm8);
  if (threadIdx.x == 0) atomicAdd(&acc[0], cnt);
  // per-image max of fg-masked align (non-negative -> uint ordering valid)
  float mv = (valid && fg) ? balign : 0.f;
#pragma unroll
  for (int off = 16; off > 0; off >>= 1) mv = fmaxf(mv, __shfl_xor(mv, off, 32));
  __syncthreads();
  if ((threadIdx.x & 31) == 0) sm8[threadIdx.x >> 5] = mv;
  __syncthreads();
  if (threadIdx.x == 0) {
    float mm = 0.f;
    for (int i = 0; i < 8; ++i) mm = fmaxf(mm, sm8[i]);
    atomicMax(&mxbits[b], __float_as_uint(mm));
  }
}

// ---------------- kernel 4: losses ----------------
__global__ void k_loss(const float* __restrict__ f0, const float* __restrict__ f1,
                       const float* __restrict__ f2,
                       const float* __restrict__ pb, const float* __restrict__ pcls,
                       const float* __restrict__ tbb, const float* __restrict__ apa,
                       const int* __restrict__ tlab, const int* __restrict__ fgm,
                       const unsigned* __restrict__ mxbits, float* __restrict__ acc) {
  int b = blockIdx.x / CHUNKS;
  int n = (blockIdx.x % CHUNKS) * 256 + threadIdx.x;
  bool valid = n < NANCH;
  __shared__ float sm8[8];
  size_t pidx = (size_t)b * NANCH + (valid ? n : 0);
  int level, pos, S; float ax, ay, st;
  anchor_of(valid ? n : 0, level, pos, S, ax, ay, st);
  float apx = ax * st, apy = ay * st;
  bool fg = valid && (fgm[pidx] != 0);
  float mx = __uint_as_float(mxbits[b]);
  float box_p = 0.f, cls_p = 0.f, dfl_p = 0.f;

  if (valid) {                     // BCE over all anchors/classes
    int tl = tlab[pidx];
    float ts_val = fg ? (apa[pidx] / (mx + 1e-9f)) : 0.f;
#pragma unroll
    for (int cc = 0; cc < NCQ; ++cc) {
      float x = pcls[pidx*NCQ + cc];
      float ts = (fg && cc == tl) ? ts_val : 0.f;
      cls_p += fmaxf(x, 0.f) - x * ts + log1pf(__expf(-fabsf(x)));
    }
  }
  if (fg) {
    float tx = tbb[pidx*4], ty = tbb[pidx*4+1], tw = tbb[pidx*4+2], th = tbb[pidx*4+3];
    float px = pb[pidx*4],  py = pb[pidx*4+1],  pw = pb[pidx*4+2],  ph = pb[pidx*4+3];
    // CIoU
    float p_x1 = px - pw*0.5f, p_y1 = py - ph*0.5f, p_x2 = px + pw*0.5f, p_y2 = py + ph*0.5f;
    float g_x1 = tx - tw*0.5f, g_y1 = ty - th*0.5f, g_x2 = tx + tw*0.5f, g_y2 = ty + th*0.5f;
    float iw = fmaxf(fminf(p_x2, g_x2) - fmaxf(p_x1, g_x1), 0.f);
    float ih = fmaxf(fminf(p_y2, g_y2) - fmaxf(p_y1, g_y1), 0.f);
    float inter = iw * ih;
    float uni = pw*ph + tw*th - inter + 1e-7f;
    float iou = inter / uni;
    float cw2 = fmaxf(p_x2, g_x2) - fminf(p_x1, g_x1);
    float ch2 = fmaxf(p_y2, g_y2) - fminf(p_y1, g_y1);
    float c2 = cw2*cw2 + ch2*ch2 + 1e-7f;
    float rho2 = (tx - px)*(tx - px) + (ty - py)*(ty - py);
    float dat = atanf(tw / (th + 1e-7f)) - atanf(pw / (ph + 1e-7f));
    float v = 0.40528473456935109f * dat * dat;   // 4/pi^2
    float al = v / (v - iou + 1.f + 1e-7f);
    box_p = 1.f - (iou - (rho2 / c2 + v * al));
    // DFL (re-read the 68 bin logits, fg anchors only; L2-resident)
    const float* f = (level == 0) ? f0 : (level == 1) ? f1 : f2;
    const float* base = f + (size_t)b * CHN * S + pos;
    float tgt4[4];
    tgt4[0] = (apx - (tx - tw*0.5f)) / st;
    tgt4[1] = (apy - (ty - th*0.5f)) / st;
    tgt4[2] = ((tx + tw*0.5f) - apx) / st;
    tgt4[3] = ((ty + th*0.5f) - apy) / st;
#pragma unroll
    for (int sd = 0; sd < 4; ++sd) {
      float t = fminf(fmaxf(tgt4[sd], 0.f), 15.99f);
      int tli = (int)floorf(t); int tri = tli + 1;
      float wl = (float)tri - t; float wr = 1.f - wl;
      float m = -1e30f;
#pragma unroll
      for (int k = 0; k < RB; ++k) m = fmaxf(m, base[(size_t)(sd*RB + k) * S]);
      float se = 0.f, vl = 0.f, vr = 0.f;
#pragma unroll
      for (int k = 0; k < RB; ++k) {
        float val = base[(size_t)(sd*RB + k) * S];
        se += __expf(val - m);
        if (k == tli) vl = val;
        if (k == tri) vr = val;
      }
      float lse = m + __logf(se);
      dfl_p -= (vl - lse) * wl + (vr - lse) * wr;
    }
  }
  float bs = block_sum256(box_p, sm8);
  if (threadIdx.x == 0) atomicAdd(&acc[1], bs);
  float cs = block_sum256(cls_p, sm8);
  if (threadIdx.x == 0) atomicAdd(&acc[2], cs);
  float ds = block_sum256(dfl_p, sm8);
  if (threadIdx.x == 0) atomicAdd(&acc[3], ds);
}

// ---------------- kernel 5: finalize ----------------
__global__ void k_final(const float* __restrict__ acc, float* __restrict__ out) {
  if (threadIdx.x == 0 && blockIdx.x == 0) {
    float np  = fmaxf(acc[0], 1.f);
    float box = acc[1] / np * 7.5f;
    float cls = acc[2] / np * 0.5f;
    float dfl = acc[3] / np * 1.5f;
    out[0] = box + cls + dfl; out[1] = box; out[2] = cls; out[3] = dfl;
  }
}

// ---------------- launch ----------------
extern "C" void kernel_launch(void* const* d_in, const int* in_sizes, int n_in,
                              void* d_out, int out_size, void* d_ws, size_t ws_size,
                              hipStream_t stream) {
  const float* f0  = (const float*)d_in[0];
  const float* f1  = (const float*)d_in[1];
  const float* f2  = (const float*)d_in[2];
  const float* tgt = (const float*)d_in[3];
  float* out = (float*)d_out;

  float* ws   = (float*)d_ws;
  float* pb   = ws;                                   // B*N*4
  float* pcls = pb   + (size_t)BQ * NANCH * 4;        // B*N*6
  float* kth  = pcls + (size_t)BQ * NANCH * NCQ;      // B*G
  float* tbb  = kth  + (size_t)BQ * GQ;               // B*N*4
  float* apa  = tbb  + (size_t)BQ * NANCH * 4;        // B*N
  int*   tlab = (int*)(apa + (size_t)BQ * NANCH);     // B*N
  int*   fgm  = tlab + (size_t)BQ * NANCH;            // B*N
  unsigned* mxbits = (unsigned*)(fgm + (size_t)BQ * NANCH); // B
  float* acc = (float*)(mxbits + BQ);                 // 4 accumulators

  // zero per-launch state (mxbits + acc are contiguous); capture-safe
  hipMemsetAsync(mxbits, 0, sizeof(unsigned) * BQ + sizeof(float) * 4, stream);

  k_predict<<<(BQ * NANCH + 255) / 256, 256, 0, stream>>>(f0, f1, f2, pb, pcls);
  k_topk   <<<BQ * GQ, 128, 0, stream>>>(tgt, pb, pcls, kth);
  k_assign <<<BQ * CHUNKS, 256, 0, stream>>>(tgt, pb, pcls, kth, tbb, apa, tlab, fgm, mxbits, acc);
  k_loss   <<<BQ * CHUNKS, 256, 0, stream>>>(f0, f1, f2, pb, pcls, tbb, apa, tlab, fgm, mxbits, acc);
  k_final  <<<1, 1, 0, stream>>>(acc, out);
}